// EGNNModel_78477642432839
// MI455X (gfx1250) — compile-verified
//
#include <hip/hip_runtime.h>
#include <math.h>

typedef __attribute__((ext_vector_type(2))) float f32x2;
typedef __attribute__((ext_vector_type(8))) float f32x8;

// Problem constants (fixed by the reference).
constexpr int NN   = 50000;   // nodes
constexpr int NE   = 800000;  // edges
constexpr int NG   = 64;      // graphs
constexpr int HID  = 128;
constexpr int EDIM = 16;
constexpr int KE   = 276;     // 2*HID + EDIM + 1 = 273, padded to mult of 4
constexpr int KE_REAL = 273;
constexpr int NLAYER  = 4;
constexpr int WPB  = 4;       // waves per block for big kernels
constexpr int MT   = 32;      // rows (M) per wave: two 16-row WMMA sub-tiles

// ---------------------------------------------------------------------------
// f32 WMMA 16x16x4 step (D = A(16x4) * B(4x16) + C).  Fragment layouts per
// CDNA5 ISA 7.12.2:
//   A: lane<16 -> {A[m][k0],A[m][k0+1]}, lane>=16 -> {A[m][k0+2],A[m][k0+3]}
//   B: lane<16 -> {B[k0][n],B[k0+1][n]}, lane>=16 -> {B[k0+2][n],B[k0+3][n]}
//   C/D: vgpr r, lane l: [M = r + 8*(l>>4)][N = l&15]
// ---------------------------------------------------------------------------
#if __has_builtin(__builtin_amdgcn_wmma_f32_16x16x4_f32)
__device__ __forceinline__ f32x8 wmma_step(f32x2 a, f32x2 b, f32x8 c, int /*lane*/) {
  return __builtin_amdgcn_wmma_f32_16x16x4_f32(false, a, false, b, (short)0, c,
                                               false, false);
}
#else
// Software emulation with identical fragment semantics (fallback only).
__device__ __forceinline__ f32x8 wmma_step(f32x2 a, f32x2 b, f32x8 c, int lane) {
  int ncol = lane & 15, mb = (lane >> 4) * 8;
  float b0 = __shfl(b.x, ncol, 32), b1 = __shfl(b.y, ncol, 32);
  float b2 = __shfl(b.x, ncol + 16, 32), b3 = __shfl(b.y, ncol + 16, 32);
#pragma unroll
  for (int r = 0; r < 8; ++r) {
    int m = mb + r;
    float a0 = __shfl(a.x, m, 32), a1 = __shfl(a.y, m, 32);
    float a2 = __shfl(a.x, m + 16, 32), a3 = __shfl(a.y, m + 16, 32);
    c[r] += a0 * b0 + a1 * b1 + a2 * b2 + a3 * b3;
  }
  return c;
}
#endif

// Per-wave GEMM, M=16: D[16][128] += A[16][4*ks] * W.
__device__ __forceinline__ void wave_gemm_f32(const float* ldsA, int rsDW,
                                              const float* __restrict__ Bp,
                                              int ksteps, int lane, f32x8 c[8]) {
  const float* aRow = ldsA + (size_t)(lane & 15) * rsDW + 2 * (lane >> 4);
  for (int s = 0; s < ksteps; ++s) {
    f32x2 a = *(const f32x2*)(aRow + 4 * s);
    const float* bp = Bp + (size_t)s * 512 + lane * 2;
#pragma unroll
    for (int t = 0; t < 8; ++t) {
      f32x2 b = *(const f32x2*)(bp + t * 64);
      c[t] = wmma_step(a, b, c[t], lane);
    }
  }
}

// Per-wave GEMM, M=32: two stacked 16-row tiles share each B fragment
// (halves L2 fragment traffic; vmem:wmma = 1:2 in steady state).
__device__ __forceinline__ void wave_gemm_f32_m32(const float* ldsA, int rsDW,
                                                  const float* __restrict__ Bp,
                                                  int ksteps, int lane,
                                                  f32x8 c0[8], f32x8 c1[8]) {
  const float* aRow0 = ldsA + (size_t)(lane & 15) * rsDW + 2 * (lane >> 4);
  const float* aRow1 = aRow0 + (size_t)16 * rsDW;
  for (int s = 0; s < ksteps; ++s) {
    f32x2 a0 = *(const f32x2*)(aRow0 + 4 * s);
    f32x2 a1 = *(const f32x2*)(aRow1 + 4 * s);
    const float* bp = Bp + (size_t)s * 512 + lane * 2;
#pragma unroll
    for (int t = 0; t < 8; ++t) {
      f32x2 b = *(const f32x2*)(bp + t * 64);
      c0[t] = wmma_step(a0, b, c0[t], lane);
      c1[t] = wmma_step(a1, b, c1[t], lane);
    }
  }
}

// ---------------------------------------------------------------------------
// Weight pre-pack: W[K][128] -> B-fragments Bp[s][t][lane][2]  (rows>=Kreal->0)
// ---------------------------------------------------------------------------
__global__ void k_pack(const float* __restrict__ W, int Kreal, int Ksteps,
                       float* __restrict__ Bp) {
  int idx = blockIdx.x * 256 + threadIdx.x;
  int tot = Ksteps * 512;
  if (idx >= tot) return;
  int p = idx & 1, lane = (idx >> 1) & 31, t = (idx >> 6) & 7, s = idx >> 9;
  int subk = ((lane >> 4) << 1) | p;
  int row = 4 * s + subk;
  int col = t * 16 + (lane & 15);
  Bp[idx] = (row < Kreal) ? W[(size_t)row * HID + col] : 0.0f;
}

// ---------------------------------------------------------------------------
// dist2[e] = ||pos[src]-pos[dst]||^2
// ---------------------------------------------------------------------------
__global__ void k_dist2(const float* __restrict__ pos, const int* __restrict__ ei,
                        float* __restrict__ dist2) {
  int e = blockIdx.x * 256 + threadIdx.x;
  if (e >= NE) return;
  int s = ei[e], d = ei[NE + e];
  float dx = pos[(size_t)s * 3 + 0] - pos[(size_t)d * 3 + 0];
  float dy = pos[(size_t)s * 3 + 1] - pos[(size_t)d * 3 + 1];
  float dz = pos[(size_t)s * 3 + 2] - pos[(size_t)d * 3 + 2];
  dist2[e] = dx * dx + dy * dy + dz * dz;
}

// ---------------------------------------------------------------------------
// Node encoder: h = relu(x @ enc_W + enc_b), x:[N,64], enc_W packed.
// ---------------------------------------------------------------------------
__global__ __launch_bounds__(256) void k_encode(const float* __restrict__ x,
                                                const float* __restrict__ Wp,
                                                const float* __restrict__ bias,
                                                float* __restrict__ h) {
  __shared__ float sX[8][16][64];
  int lane = threadIdx.x & 31, w = threadIdx.x >> 5;
  int tile = blockIdx.x * 8 + w;
  if (tile >= NN / 16) return;
  int n0 = tile * 16;
  for (int r = 0; r < 16; ++r) {
    f32x2 v = *(const f32x2*)(x + (size_t)(n0 + r) * 64 + lane * 2);
    *(f32x2*)&sX[w][r][lane * 2] = v;
  }
  f32x8 c[8];
#pragma unroll
  for (int t = 0; t < 8; ++t) c[t] = (f32x8)(0.0f);
  wave_gemm_f32(&sX[w][0][0], 64, Wp, 16, lane, c);
  int ncol = lane & 15, mb = (lane >> 4) * 8;
#pragma unroll
  for (int t = 0; t < 8; ++t) {
    int col = t * 16 + ncol;
    float b = bias[col];
#pragma unroll
    for (int r = 0; r < 8; ++r)
      h[(size_t)(n0 + mb + r) * HID + col] = fmaxf(c[t][r] + b, 0.0f);
  }
}

// ---------------------------------------------------------------------------
// Edge MLP + scatter:  m = relu(relu(feat@W1+b1)@W2+b2); agg[dst] += m
// feat = [h[dst] | h[src] | edge_attr | dist2] (K=273, padded to 276).
// One 32-edge tile per wave; ReLU intermediate aliased into feat cols 0..127.
// ---------------------------------------------------------------------------
__global__ __launch_bounds__(128) void k_edge_mlp(
    const float* __restrict__ h, const int* __restrict__ ei,
    const float* __restrict__ edge_attr, const float* __restrict__ dist2,
    const float* __restrict__ W1p, const float* __restrict__ b1,
    const float* __restrict__ W2p, const float* __restrict__ b2,
    float* __restrict__ agg) {
  __shared__ float sFeat[WPB][MT][KE];
  __shared__ int sDst[WPB][MT];
  int lane = threadIdx.x & 31, w = threadIdx.x >> 5;
  int tile = blockIdx.x * WPB + w;
  if (tile >= NE / MT) return;
  int e0 = tile * MT;

  for (int r = 0; r < MT; ++r) {
    int e = e0 + r;
    int di = ei[NE + e];
    int sj = ei[e];
    if (lane == 0) sDst[w][r] = di;
    float4 hd = *(const float4*)(h + (size_t)di * HID + lane * 4);
    *(float4*)&sFeat[w][r][lane * 4] = hd;
    float4 hs = *(const float4*)(h + (size_t)sj * HID + lane * 4);
    *(float4*)&sFeat[w][r][HID + lane * 4] = hs;
    if (lane < EDIM) sFeat[w][r][2 * HID + lane] = edge_attr[(size_t)e * EDIM + lane];
    if (lane < 4) sFeat[w][r][272 + lane] = (lane == 0) ? dist2[e] : 0.0f;
  }

  f32x8 c0[8], c1[8];
#pragma unroll
  for (int t = 0; t < 8; ++t) { c0[t] = (f32x8)(0.0f); c1[t] = (f32x8)(0.0f); }
  wave_gemm_f32_m32(&sFeat[w][0][0], KE, W1p, KE / 4, lane, c0, c1);

  int ncol = lane & 15, mb = (lane >> 4) * 8;
  // relu + bias -> write intermediate into feat cols 0..127 (dead after GEMM1)
#pragma unroll
  for (int t = 0; t < 8; ++t) {
    int col = t * 16 + ncol;
    float b = b1[col];
#pragma unroll
    for (int r = 0; r < 8; ++r)
      sFeat[w][mb + r][col] = fmaxf(c0[t][r] + b, 0.0f);
#pragma unroll
    for (int r = 0; r < 8; ++r)
      sFeat[w][16 + mb + r][col] = fmaxf(c1[t][r] + b, 0.0f);
  }

#pragma unroll
  for (int t = 0; t < 8; ++t) { c0[t] = (f32x8)(0.0f); c1[t] = (f32x8)(0.0f); }
  wave_gemm_f32_m32(&sFeat[w][0][0], KE, W2p, HID / 4, lane, c0, c1);

  int nodeIdx0[8], nodeIdx1[8];
#pragma unroll
  for (int r = 0; r < 8; ++r) {
    nodeIdx0[r] = sDst[w][mb + r];
    nodeIdx1[r] = sDst[w][16 + mb + r];
  }
#pragma unroll
  for (int t = 0; t < 8; ++t) {
    int col = t * 16 + ncol;
    float b = b2[col];
#pragma unroll
    for (int r = 0; r < 8; ++r) {
      float v = fmaxf(c0[t][r] + b, 0.0f);
      atomicAdd(&agg[(size_t)nodeIdx0[r] * HID + col], v);
    }
#pragma unroll
    for (int r = 0; r < 8; ++r) {
      float v = fmaxf(c1[t][r] + b, 0.0f);
      atomicAdd(&agg[(size_t)nodeIdx1[r] * HID + col], v);
    }
  }
}

// ---------------------------------------------------------------------------
// Node MLP:  hOut = relu([h|agg]@W1+b1)@W2+b2   (no trailing relu)
// One 32-node tile per wave; intermediate aliased into feat cols 0..127.
// ---------------------------------------------------------------------------
__global__ __launch_bounds__(128) void k_node_mlp(
    const float* __restrict__ h, const float* __restrict__ agg,
    const float* __restrict__ W1p, const float* __restrict__ b1,
    const float* __restrict__ W2p, const float* __restrict__ b2,
    float* __restrict__ hOut) {
  __shared__ float sFeat[WPB][MT][2 * HID];
  int lane = threadIdx.x & 31, w = threadIdx.x >> 5;
  constexpr int NT32 = (NN + MT - 1) / MT;  // 1563 (last tile partial)
  int tile = blockIdx.x * WPB + w;
  if (tile >= NT32) return;
  int n0 = tile * MT;

  for (int r = 0; r < MT; ++r) {
    int n = n0 + r;
    if (n >= NN) n = NN - 1;  // clamp (safe read; stores are guarded)
    float4 hv = *(const float4*)(h + (size_t)n * HID + lane * 4);
    *(float4*)&sFeat[w][r][lane * 4] = hv;
    float4 av = *(const float4*)(agg + (size_t)n * HID + lane * 4);
    *(float4*)&sFeat[w][r][HID + lane * 4] = av;
  }

  f32x8 c0[8], c1[8];
#pragma unroll
  for (int t = 0; t < 8; ++t) { c0[t] = (f32x8)(0.0f); c1[t] = (f32x8)(0.0f); }
  wave_gemm_f32_m32(&sFeat[w][0][0], 2 * HID, W1p, (2 * HID) / 4, lane, c0, c1);

  int ncol = lane & 15, mb = (lane >> 4) * 8;
#pragma unroll
  for (int t = 0; t < 8; ++t) {
    int col = t * 16 + ncol;
    float b = b1[col];
#pragma unroll
    for (int r = 0; r < 8; ++r)
      sFeat[w][mb + r][col] = fmaxf(c0[t][r] + b, 0.0f);
#pragma unroll
    for (int r = 0; r < 8; ++r)
      sFeat[w][16 + mb + r][col] = fmaxf(c1[t][r] + b, 0.0f);
  }

#pragma unroll
  for (int t = 0; t < 8; ++t) { c0[t] = (f32x8)(0.0f); c1[t] = (f32x8)(0.0f); }
  wave_gemm_f32_m32(&sFeat[w][0][0], 2 * HID, W2p, HID / 4, lane, c0, c1);

#pragma unroll
  for (int t = 0; t < 8; ++t) {
    int col = t * 16 + ncol;
    float b = b2[col];
#pragma unroll
    for (int r = 0; r < 8; ++r) {
      int gr = n0 + mb + r;
      if (gr < NN) hOut[(size_t)gr * HID + col] = c0[t][r] + b;  // no relu
    }
#pragma unroll
    for (int r = 0; r < 8; ++r) {
      int gr = n0 + 16 + mb + r;
      if (gr < NN) hOut[(size_t)gr * HID + col] = c1[t][r] + b;  // no relu
    }
  }
}

// ---------------------------------------------------------------------------
// Pooling: segment sum / (order-preserving-key) max / count
// ---------------------------------------------------------------------------
__global__ void k_pool_init(float* ssum, unsigned* smaxk, float* cnt) {
  int i = blockIdx.x * 256 + threadIdx.x;
  if (i < NG * HID) { ssum[i] = 0.0f; smaxk[i] = 0x007FFFFFu; }  // ~(-inf bits)
  if (i < NG) cnt[i] = 0.0f;
}

__global__ void k_pool(const float* __restrict__ h, const int* __restrict__ batch,
                       float* __restrict__ ssum, unsigned* __restrict__ smaxk,
                       float* __restrict__ cnt) {
  size_t idx = (size_t)blockIdx.x * 256 + threadIdx.x;
  if (idx >= (size_t)NN * HID) return;
  int n = (int)(idx / HID), c = (int)(idx % HID);
  int g = batch[n];
  float v = h[idx];
  atomicAdd(&ssum[(size_t)g * HID + c], v);
  unsigned bits = __float_as_uint(v);
  unsigned key = (bits & 0x80000000u) ? ~bits : (bits | 0x80000000u);
  atomicMax(&smaxk[(size_t)g * HID + c], key);
  if (c == 0) atomicAdd(&cnt[g], 1.0f);
}

// ---------------------------------------------------------------------------
// Readout MLP (tiny): out = softplus(relu(relu(pooled@rW1+b)@rW2+b)@rW3+b)
// ---------------------------------------------------------------------------
__global__ __launch_bounds__(256) void k_readout(
    const float* __restrict__ ssum, const unsigned* __restrict__ smaxk,
    const float* __restrict__ cnt, const float* __restrict__ rW1,
    const float* __restrict__ rb1, const float* __restrict__ rW2,
    const float* __restrict__ rb2, const float* __restrict__ rW3,
    const float* __restrict__ rb3, float* __restrict__ out) {
  __shared__ float pooled[3 * HID];
  __shared__ float o1[HID];
  __shared__ float o2[HID / 2];
  int t = threadIdx.x;
  for (int g = 0; g < NG; ++g) {
    if (t < HID) {
      float s = ssum[(size_t)g * HID + t];
      float c = fmaxf(cnt[g], 1.0f);
      unsigned k = smaxk[(size_t)g * HID + t];
      unsigned bits = (k & 0x80000000u) ? (k ^ 0x80000000u) : ~k;
      pooled[t] = s / c;
      pooled[HID + t] = __uint_as_float(bits);
      pooled[2 * HID + t] = s;
    }
    __syncthreads();
    if (t < HID) {
      float acc = rb1[t];
      for (int k = 0; k < 3 * HID; ++k) acc += pooled[k] * rW1[(size_t)k * HID + t];
      o1[t] = fmaxf(acc, 0.0f);
    }
    __syncthreads();
    if (t < HID / 2) {
      float acc = rb2[t];
      for (int k = 0; k < HID; ++k) acc += o1[k] * rW2[(size_t)k * (HID / 2) + t];
      o2[t] = fmaxf(acc, 0.0f);
    }
    __syncthreads();
    if (t == 0) {
      float acc = rb3[0];
      for (int k = 0; k < HID / 2; ++k) acc += o2[k] * rW3[k];
      out[g] = (acc > 20.0f) ? acc : log1pf(expf(acc));  // softplus
    }
    __syncthreads();
  }
}

// ---------------------------------------------------------------------------
extern "C" void kernel_launch(void* const* d_in, const int* in_sizes, int n_in,
                              void* d_out, int out_size, void* d_ws, size_t ws_size,
                              hipStream_t stream) {
  (void)in_sizes; (void)n_in; (void)out_size; (void)ws_size;
  const float* x         = (const float*)d_in[0];
  const float* pos       = (const float*)d_in[1];
  const float* edge_attr = (const float*)d_in[2];
  const int*   ei        = (const int*)d_in[3];
  const int*   batch     = (const int*)d_in[4];
  const float* enc_W     = (const float*)d_in[5];
  const float* enc_b     = (const float*)d_in[6];
  const float* eW1       = (const float*)d_in[7];
  const float* eb1       = (const float*)d_in[8];
  const float* eW2       = (const float*)d_in[9];
  const float* eb2       = (const float*)d_in[10];
  const float* nW1       = (const float*)d_in[11];
  const float* nb1       = (const float*)d_in[12];
  const float* nW2       = (const float*)d_in[13];
  const float* nb2       = (const float*)d_in[14];
  const float* rW1       = (const float*)d_in[15];
  const float* rb1       = (const float*)d_in[16];
  const float* rW2       = (const float*)d_in[17];
  const float* rb2       = (const float*)d_in[18];
  const float* rW3       = (const float*)d_in[19];
  const float* rb3       = (const float*)d_in[20];

  char* ws = (char*)d_ws;
  size_t off = 0;
  auto alloc = [&](size_t bytes) -> void* {
    void* p = (void*)(ws + off);
    off += (bytes + 255) & ~(size_t)255;
    return p;
  };
  float*    hA    = (float*)alloc((size_t)NN * HID * 4);
  float*    hB    = (float*)alloc((size_t)NN * HID * 4);
  float*    agg   = (float*)alloc((size_t)NN * HID * 4);
  float*    dist2 = (float*)alloc((size_t)NE * 4);
  float*    encP  = (float*)alloc((size_t)16 * 512 * 4);
  float*    eW1P  = (float*)alloc((size_t)NLAYER * 69 * 512 * 4);
  float*    eW2P  = (float*)alloc((size_t)NLAYER * 32 * 512 * 4);
  float*    nW1P  = (float*)alloc((size_t)NLAYER * 64 * 512 * 4);
  float*    nW2P  = (float*)alloc((size_t)NLAYER * 32 * 512 * 4);
  float*    ssum  = (float*)alloc((size_t)NG * HID * 4);
  unsigned* smaxk = (unsigned*)alloc((size_t)NG * HID * 4);
  float*    cnt   = (float*)alloc((size_t)NG * 4);

  // ---- pack all weights into WMMA B-fragment layout ----
  k_pack<<<(16 * 512 + 255) / 256, 256, 0, stream>>>(enc_W, 64, 16, encP);
  for (int l = 0; l < NLAYER; ++l) {
    k_pack<<<(69 * 512 + 255) / 256, 256, 0, stream>>>(
        eW1 + (size_t)l * KE_REAL * HID, KE_REAL, 69, eW1P + (size_t)l * 69 * 512);
    k_pack<<<(32 * 512 + 255) / 256, 256, 0, stream>>>(
        eW2 + (size_t)l * HID * HID, HID, 32, eW2P + (size_t)l * 32 * 512);
    k_pack<<<(64 * 512 + 255) / 256, 256, 0, stream>>>(
        nW1 + (size_t)l * 2 * HID * HID, 2 * HID, 64, nW1P + (size_t)l * 64 * 512);
    k_pack<<<(32 * 512 + 255) / 256, 256, 0, stream>>>(
        nW2 + (size_t)l * HID * HID, HID, 32, nW2P + (size_t)l * 32 * 512);
  }

  // ---- geometry + encoder ----
  k_dist2<<<(NE + 255) / 256, 256, 0, stream>>>(pos, ei, dist2);
  k_encode<<<(NN / 16 + 7) / 8, 256, 0, stream>>>(x, encP, enc_b, hA);

  // ---- message-passing layers ----
  float* hin = hA;
  float* hout = hB;
  constexpr int NT32_N = (NN + MT - 1) / MT;
  for (int l = 0; l < NLAYER; ++l) {
    hipMemsetAsync(agg, 0, (size_t)NN * HID * 4, stream);
    k_edge_mlp<<<(NE / MT + WPB - 1) / WPB, 32 * WPB, 0, stream>>>(
        hin, ei, edge_attr, dist2,
        eW1P + (size_t)l * 69 * 512, eb1 + (size_t)l * HID,
        eW2P + (size_t)l * 32 * 512, eb2 + (size_t)l * HID, agg);
    k_node_mlp<<<(NT32_N + WPB - 1) / WPB, 32 * WPB, 0, stream>>>(
        hin, agg,
        nW1P + (size_t)l * 64 * 512, nb1 + (size_t)l * HID,
        nW2P + (size_t)l * 32 * 512, nb2 + (size_t)l * HID, hout);
    float* tmp = hin; hin = hout; hout = tmp;
  }

  // ---- pooling + readout ----
  k_pool_init<<<(NG * HID + 255) / 256, 256, 0, stream>>>(ssum, smaxk, cnt);
  k_pool<<<(int)(((size_t)NN * HID + 255) / 256), 256, 0, stream>>>(hin, batch,
                                                                    ssum, smaxk, cnt);
  k_readout<<<1, 256, 0, stream>>>(ssum, smaxk, cnt, rW1, rb1, rW2, rb2, rW3, rb3,
                                   (float*)d_out);
}